// SelfAttention_10488310137271
// MI455X (gfx1250) — compile-verified
//
#include <hip/hip_runtime.h>

typedef float v2f __attribute__((ext_vector_type(2)));
typedef float v4f __attribute__((ext_vector_type(4)));
typedef float v8f __attribute__((ext_vector_type(8)));

#define LSEQ 2048
#define EDIM 1024
#define NH   16
#define HD   64
#define INV_SQRT_E 0.03125f  // 1/sqrt(1024)
#define KSTRIDE 68           // padded LDS row stride for K/V tiles

// D = A(16x4) * B(4x16) + C, fp32. 8-arg form:
// (neg_a, A, neg_b, B, c_mod, C, reuse_a, reuse_b)
__device__ __forceinline__ v8f wmma_f32(v2f a, v2f b, v8f c) {
  return __builtin_amdgcn_wmma_f32_16x16x4_f32(false, a, false, b, (short)0, c,
                                               false, false);
}

// ---------------------------------------------------------------------------
// NT GEMM: Out[m, n] = sum_k X[m, k] * W[n, k] (+ bias[n])
// One wave computes a 64x64 tile: 4x4 grid of 16x16 WMMA accumulators.
// A-frag (16x4 fp32): lanes 0-15 hold M=lane, K = {0,1}; lanes 16-31 hold
// M=lane-16, K = {2,3}  -> per-lane contiguous float2 at X[row, kb + 2*hi].
// B-frag mirrors A with N in place of M (W is read row-wise: B[k][n]=W[n,k]).
// C/D: VGPR r, lanes 0-15 -> (M=r, N=lane); lanes 16-31 -> (M=r+8, N=lane-16).
// ---------------------------------------------------------------------------
__device__ __forceinline__ void gemm_nt_64x64(const float* __restrict__ X,
                                              const float* __restrict__ W,
                                              const float* __restrict__ bias,
                                              float* __restrict__ Out) {
  const int lane = threadIdx.x;
  const int lo = lane & 15;
  const int hi = lane >> 4;
  const int row0 = blockIdx.x * 64;
  const int col0 = blockIdx.y * 64;

  v8f acc[4][4];
#pragma unroll
  for (int i = 0; i < 4; ++i)
#pragma unroll
    for (int j = 0; j < 4; ++j)
      acc[i][j] = {0.f, 0.f, 0.f, 0.f, 0.f, 0.f, 0.f, 0.f};

  for (int kb = 0; kb < EDIM; kb += 4) {
    v2f a[4], b[4];
#pragma unroll
    for (int t = 0; t < 4; ++t) {
      a[t] = *(const v2f*)(X + (size_t)(row0 + t * 16 + lo) * EDIM + kb + 2 * hi);
      b[t] = *(const v2f*)(W + (size_t)(col0 + t * 16 + lo) * EDIM + kb + 2 * hi);
    }
#pragma unroll
    for (int i = 0; i < 4; ++i)
#pragma unroll
      for (int j = 0; j < 4; ++j) acc[i][j] = wmma_f32(a[i], b[j], acc[i][j]);
  }

#pragma unroll
  for (int i = 0; i < 4; ++i)
#pragma unroll
    for (int j = 0; j < 4; ++j) {
      const int col = col0 + j * 16 + lo;
      const float bv = bias ? bias[col] : 0.0f;
#pragma unroll
      for (int r = 0; r < 8; ++r) {
        const int row = row0 + i * 16 + r + 8 * hi;
        Out[(size_t)row * EDIM + col] = acc[i][j][r] + bv;
      }
    }
}

__global__ __launch_bounds__(32) void proj3(
    const float* __restrict__ values, const float* __restrict__ keys,
    const float* __restrict__ query, const float* __restrict__ Wv,
    const float* __restrict__ Wk, const float* __restrict__ Wq,
    float* __restrict__ vout, float* __restrict__ kout,
    float* __restrict__ qout) {
  const float* X;
  const float* W;
  float* O;
  if (blockIdx.z == 0) {
    X = values; W = Wv; O = vout;
  } else if (blockIdx.z == 1) {
    X = keys; W = Wk; O = kout;
  } else {
    X = query; W = Wq; O = qout;
  }
  gemm_nt_64x64(X, W, nullptr, O);
}

__global__ __launch_bounds__(32) void gemm_out(const float* __restrict__ X,
                                               const float* __restrict__ W,
                                               const float* __restrict__ bias,
                                               float* __restrict__ Out) {
  gemm_nt_64x64(X, W, bias, Out);
}

// ---------------------------------------------------------------------------
// Fused attention with distance bias, flash-style online softmax.
// 4 waves / block; each wave owns 16 query rows (block covers 64).
// K/V 16x64 tiles are double-buffered in LDS, shared by all 4 waves, with
// the next tile's global loads overlapped with compute on the current tile.
// ---------------------------------------------------------------------------
__global__ __launch_bounds__(128) void attn_fused(
    const float* __restrict__ Q, const float* __restrict__ K,
    const float* __restrict__ V, const float* __restrict__ dist,
    float* __restrict__ Oout) {
  __shared__ float kbuf[2][16 * KSTRIDE];
  __shared__ float vbuf[2][16 * KSTRIDE];
  __shared__ float pbuf[4][16 * 18];  // per-wave padded P tile (C->A transpose)

  const int tid = threadIdx.x;
  const int lane = tid & 31;
  const int w = tid >> 5;  // wave id 0..3
  const int lo = lane & 15;
  const int hi = lane >> 4;
  const int qb = blockIdx.x * 64 + w * 16;
  const int h = blockIdx.y;
  const size_t hoff = (size_t)h * HD;

  // cooperative staging indices: 128 threads x 8 floats = one 16x64 tile
  const int srow = tid >> 3;       // 0..15
  const int scol = (tid & 7) * 8;  // 0,8,...,56

  // Q tile as 16 A-fragments over D=64 (k-steps of 4)
  v2f qa[16];
#pragma unroll
  for (int d4 = 0; d4 < 16; ++d4)
    qa[d4] = *(const v2f*)(Q + (size_t)(qb + lo) * EDIM + hoff + 4 * d4 + 2 * hi);

  float mrow[8], srun[8];
  v8f o[4];
#pragma unroll
  for (int r = 0; r < 8; ++r) {
    mrow[r] = -3.0e38f;
    srun[r] = 0.0f;
  }
#pragma unroll
  for (int t = 0; t < 4; ++t) o[t] = {0.f, 0.f, 0.f, 0.f, 0.f, 0.f, 0.f, 0.f};

  // prologue: stage key/value tile 0 into buffer 0
  {
    const float* kg = K + (size_t)srow * EDIM + hoff + scol;
    const float* vg = V + (size_t)srow * EDIM + hoff + scol;
    v4f k0 = *(const v4f*)kg;
    v4f k1 = *(const v4f*)(kg + 4);
    v4f v0 = *(const v4f*)vg;
    v4f v1 = *(const v4f*)(vg + 4);
    *(v4f*)&kbuf[0][srow * KSTRIDE + scol] = k0;
    *(v4f*)&kbuf[0][srow * KSTRIDE + scol + 4] = k1;
    *(v4f*)&vbuf[0][srow * KSTRIDE + scol] = v0;
    *(v4f*)&vbuf[0][srow * KSTRIDE + scol + 4] = v1;
  }
  __syncthreads();

  const int NT = LSEQ / 16;
  for (int kt = 0; kt < NT; ++kt) {
    const int kb = kt * 16;
    const int cur = kt & 1;
    const int nxt = cur ^ 1;
    const bool havenext = (kt + 1) < NT;  // block-uniform

    // issue next tile's global loads early (overlap with WMMA compute)
    v4f pk0, pk1, pv0, pv1;
    if (havenext) {
      const float* kg = K + (size_t)(kb + 16 + srow) * EDIM + hoff + scol;
      const float* vg = V + (size_t)(kb + 16 + srow) * EDIM + hoff + scol;
      pk0 = *(const v4f*)kg;
      pk1 = *(const v4f*)(kg + 4);
      pv0 = *(const v4f*)vg;
      pv1 = *(const v4f*)(vg + 4);
    }

    const float* klds = kbuf[cur];
    const float* vlds = vbuf[cur];

    // energy tile: E = Q * K^T  (B[k=d][n=key] = K[kb+n, d], from LDS)
    v8f e = {0.f, 0.f, 0.f, 0.f, 0.f, 0.f, 0.f, 0.f};
#pragma unroll
    for (int d4 = 0; d4 < 16; ++d4) {
      v2f kf = *(const v2f*)(klds + lo * KSTRIDE + 4 * d4 + 2 * hi);
      e = wmma_f32(qa[d4], kf, e);
    }

    // distance bias + scale + online softmax (per C-layout row r)
    float p[8];
#pragma unroll
    for (int r = 0; r < 8; ++r) {
      const int row = qb + r + 8 * hi;
      const float dm = dist[(size_t)row * LSEQ + kb + lo];
      const float bias = (dm == 0.0f) ? 0.0f : 1.0f / (dm + 1.0f);
      const float z = (e[r] + bias) * INV_SQRT_E;

      float mx = z;  // row max across the 16 lanes holding this row
      mx = fmaxf(mx, __shfl_xor(mx, 1, 32));
      mx = fmaxf(mx, __shfl_xor(mx, 2, 32));
      mx = fmaxf(mx, __shfl_xor(mx, 4, 32));
      mx = fmaxf(mx, __shfl_xor(mx, 8, 32));
      const float mnew = fmaxf(mrow[r], mx);
      const float c = __expf(mrow[r] - mnew);
      const float pe = __expf(z - mnew);
      float rs = pe;
      rs += __shfl_xor(rs, 1, 32);
      rs += __shfl_xor(rs, 2, 32);
      rs += __shfl_xor(rs, 4, 32);
      rs += __shfl_xor(rs, 8, 32);
      srun[r] = srun[r] * c + rs;
      mrow[r] = mnew;
      p[r] = pe;
#pragma unroll
      for (int nt = 0; nt < 4; ++nt) o[nt][r] *= c;  // rescale O row
    }

    // transpose P (C layout -> row-major LDS -> A frags); wave-private slice,
    // same-wave LDS is in-order, compiler inserts the dscnt wait for reads.
    float* pw = pbuf[w];
#pragma unroll
    for (int r = 0; r < 8; ++r) pw[(r + 8 * hi) * 18 + lo] = p[r];
    asm volatile("" ::: "memory");
    v2f pa[4];
#pragma unroll
    for (int kc = 0; kc < 4; ++kc)
      pa[kc] = *(const v2f*)(pw + lo * 18 + 4 * kc + 2 * hi);

    // O += P * V  (B[k=key][n=d] = V[kb+key, nt*16 + n], from LDS)
#pragma unroll
    for (int kc = 0; kc < 4; ++kc) {
      const float* vrow = vlds + (4 * kc + 2 * hi) * KSTRIDE;
#pragma unroll
      for (int nt = 0; nt < 4; ++nt) {
        v2f vb;
        vb.x = vrow[nt * 16 + lo];
        vb.y = vrow[KSTRIDE + nt * 16 + lo];
        o[nt] = wmma_f32(pa[kc], vb, o[nt]);
      }
    }

    // commit next tile into the other buffer, then block barrier
    if (havenext) {
      *(v4f*)&kbuf[nxt][srow * KSTRIDE + scol] = pk0;
      *(v4f*)&kbuf[nxt][srow * KSTRIDE + scol + 4] = pk1;
      *(v4f*)&vbuf[nxt][srow * KSTRIDE + scol] = pv0;
      *(v4f*)&vbuf[nxt][srow * KSTRIDE + scol + 4] = pv1;
    }
    __syncthreads();
  }

  // normalize and write [L, H*D]
#pragma unroll
  for (int nt = 0; nt < 4; ++nt)
#pragma unroll
    for (int r = 0; r < 8; ++r) {
      const int row = qb + r + 8 * hi;
      Oout[(size_t)row * EDIM + hoff + nt * 16 + lo] = o[nt][r] / srun[r];
    }
}

// ---------------------------------------------------------------------------
extern "C" void kernel_launch(void* const* d_in, const int* in_sizes, int n_in,
                              void* d_out, int out_size, void* d_ws,
                              size_t ws_size, hipStream_t stream) {
  const float* values = (const float*)d_in[0];
  const float* keys = (const float*)d_in[1];
  const float* query = (const float*)d_in[2];
  const float* dist = (const float*)d_in[3];
  const float* Wv = (const float*)d_in[4];
  const float* Wk = (const float*)d_in[5];
  const float* Wq = (const float*)d_in[6];
  const float* Wo = (const float*)d_in[7];
  const float* bo = (const float*)d_in[8];
  float* out = (float*)d_out;

  float* vproj = (float*)d_ws;
  float* kproj = vproj + (size_t)LSEQ * EDIM;
  float* qproj = kproj + (size_t)LSEQ * EDIM;
  float* aout = qproj + (size_t)LSEQ * EDIM;

  // 1) three input projections (z selects values/keys/query)
  dim3 pgrid(LSEQ / 64, EDIM / 64, 3);
  proj3<<<pgrid, dim3(32), 0, stream>>>(values, keys, query, Wv, Wk, Wq, vproj,
                                        kproj, qproj);

  // 2) fused biased attention: 4 waves/block, 64 query rows/block, per head
  dim3 agrid(LSEQ / 64, NH);
  attn_fused<<<agrid, dim3(128), 0, stream>>>(qproj, kproj, vproj, dist, aout);

  // 3) output projection + bias
  dim3 ogrid(LSEQ / 64, EDIM / 64, 1);
  gemm_out<<<ogrid, dim3(32), 0, stream>>>(aout, Wo, bo, out);
}